// KGAT_532575945232
// MI455X (gfx1250) — compile-verified
//
#include <hip/hip_runtime.h>
#include <hip/hip_bf16.h>
#include <math.h>

#define N_NODES 50000
#define N_EDGES 600000
#define N_REL   64
#define DIM     128
#define N_LAYERS 2

typedef float __attribute__((ext_vector_type(2))) v2f;
typedef float __attribute__((ext_vector_type(8))) v8f;

// ---------- helpers ----------

__device__ __forceinline__ unsigned orderEnc(float f) {
    unsigned u = __float_as_uint(f);
    return (u & 0x80000000u) ? ~u : (u | 0x80000000u);
}
__device__ __forceinline__ float orderDec(unsigned u) {
    unsigned b = (u & 0x80000000u) ? (u & 0x7FFFFFFFu) : ~u;
    return __uint_as_float(b);
}

__global__ void fill_u32_kernel(unsigned* p, unsigned v, int n) {
    int i = blockIdx.x * blockDim.x + threadIdx.x;
    if (i < n) p[i] = v;
}

// ---------- 1. degree ----------
__global__ void deg_kernel(const int* __restrict__ edst, int* __restrict__ deg, int E) {
    int e = blockIdx.x * blockDim.x + threadIdx.x;
    if (e < E) atomicAdd(&deg[edst[e]], 1);
}

// ---------- 2. per-edge score: dot(prev[nbr], tanh(ent[dst]+rel[r])) ----------
__global__ void score_kernel(const float* __restrict__ prev,
                             const float* __restrict__ ent,
                             const float* __restrict__ rel,
                             const int* __restrict__ edst,
                             const int* __restrict__ enbr,
                             const int* __restrict__ erel,
                             float* __restrict__ score, int E) {
    int wave = (blockIdx.x * blockDim.x + threadIdx.x) >> 5;
    int lane = threadIdx.x & 31;
    if (wave >= E) return;
    int dst = edst[wave], nbr = enbr[wave], r = erel[wave];
    const float4* ed = (const float4*)(ent + (size_t)dst * DIM);
    const float4* rr = (const float4*)(rel + (size_t)r   * DIM);
    const float4* nb = (const float4*)(prev + (size_t)nbr * DIM);
    float4 e4 = ed[lane], r4 = rr[lane], n4 = nb[lane];
    float s = n4.x * tanhf(e4.x + r4.x)
            + n4.y * tanhf(e4.y + r4.y)
            + n4.z * tanhf(e4.z + r4.z)
            + n4.w * tanhf(e4.w + r4.w);
    #pragma unroll
    for (int off = 16; off; off >>= 1) s += __shfl_xor(s, off, 32);
    if (lane == 0) score[wave] = s;
}

// ---------- 3. segment max (order-encoded uint atomicMax) ----------
__global__ void smax_kernel(const float* __restrict__ score,
                            const int* __restrict__ edst,
                            unsigned* __restrict__ smax_u, int E) {
    int e = blockIdx.x * blockDim.x + threadIdx.x;
    if (e < E) atomicMax(&smax_u[edst[e]], orderEnc(score[e]));
}

// ---------- 4. ex = exp(score - max); denom += ex ----------
__global__ void exdenom_kernel(float* __restrict__ score,
                               const int* __restrict__ edst,
                               const unsigned* __restrict__ smax_u,
                               float* __restrict__ denom, int E) {
    int e = blockIdx.x * blockDim.x + threadIdx.x;
    if (e >= E) return;
    int dst = edst[e];
    float ex = expf(score[e] - orderDec(smax_u[dst]));
    score[e] = ex;
    atomicAdd(&denom[dst], ex);
}

// ---------- 5. neigh[dst] += alpha * prev[nbr] ----------
__global__ void scatter_kernel(const float* __restrict__ ex,
                               const float* __restrict__ denom,
                               const float* __restrict__ prev,
                               const int* __restrict__ edst,
                               const int* __restrict__ enbr,
                               float* __restrict__ neigh, int E) {
    int wave = (blockIdx.x * blockDim.x + threadIdx.x) >> 5;
    int lane = threadIdx.x & 31;
    if (wave >= E) return;
    int dst = edst[wave], nbr = enbr[wave];
    float alpha = ex[wave] / denom[dst];
    float4 n4 = ((const float4*)(prev + (size_t)nbr * DIM))[lane];
    float* d = neigh + (size_t)dst * DIM + lane * 4;
    atomicAdd(d + 0, alpha * n4.x);
    atomicAdd(d + 1, alpha * n4.y);
    atomicAdd(d + 2, alpha * n4.z);
    atomicAdd(d + 3, alpha * n4.w);
}

// ---------- 6. WMMA GEMM: h = leakyrelu(concat(prev,neigh) @ Wl + bl) ----------
// One wave per 16x16 output tile. K = 256 in steps of 4 using
// V_WMMA_F32_16X16X4_F32 (fp32 in, fp32 accum).
// A layout (16x4 f32): lanes 0-15 -> M=lane,  V0=K0 V1=K1
//                      lanes 16-31-> M=lane-16,V0=K2 V1=K3
// B layout (4x16 f32): symmetric with N across lanes.
// C/D layout: VGPR v, lanes 0-15 -> row v, col lane; lanes 16-31 -> row v+8.
__global__ void gemm_kernel(const float* __restrict__ prev,
                            const float* __restrict__ neigh,
                            const float* __restrict__ Wl,   // [256,128]
                            const float* __restrict__ bl,   // [128]
                            float* __restrict__ h) {
    int wave  = (blockIdx.x * blockDim.x + threadIdx.x) >> 5;
    int lane  = threadIdx.x & 31;
    int mtile = wave >> 3;            // 0..3124  (50000/16 exact)
    int ntile = wave & 7;             // 0..7     (128/16 exact)
    int m0 = mtile << 4, n0 = ntile << 4;
    int half = lane >> 4, lid = lane & 15;
    const float* arow0 = prev  + (size_t)(m0 + lid) * DIM;   // first 128 of K
    const float* arow1 = neigh + (size_t)(m0 + lid) * DIM;   // last 128 of K
    int bcol = n0 + lid;

    v8f c = {};
    #pragma unroll
    for (int k0 = 0; k0 < 128; k0 += 4) {
        int ka = k0 + 2 * half;
        v2f a, b;
        a.x = arow0[ka];
        a.y = arow0[ka + 1];
        b.x = Wl[(size_t)ka * DIM + bcol];
        b.y = Wl[(size_t)(ka + 1) * DIM + bcol];
        c = __builtin_amdgcn_wmma_f32_16x16x4_f32(false, a, false, b,
                                                  (short)0, c, false, false);
    }
    #pragma unroll
    for (int k0 = 0; k0 < 128; k0 += 4) {
        int ka = k0 + 2 * half;
        v2f a, b;
        a.x = arow1[ka];
        a.y = arow1[ka + 1];
        b.x = Wl[(size_t)(128 + ka) * DIM + bcol];
        b.y = Wl[(size_t)(128 + ka + 1) * DIM + bcol];
        c = __builtin_amdgcn_wmma_f32_16x16x4_f32(false, a, false, b,
                                                  (short)0, c, false, false);
    }

    float bias = bl[bcol];
    #pragma unroll
    for (int v = 0; v < 8; ++v) {
        int row = m0 + v + 8 * half;
        float x = c[v] + bias;
        x = x > 0.0f ? x : 0.01f * x;      // LeakyReLU slope 0.01
        h[(size_t)row * DIM + bcol] = x;
    }
}

// ---------- 7. row L2-normalize + has_nbr fallback ----------
__global__ void norm_kernel(const float* __restrict__ h,
                            const int* __restrict__ deg,
                            const float* __restrict__ ent,
                            float* __restrict__ out, int N) {
    int wave = (blockIdx.x * blockDim.x + threadIdx.x) >> 5;
    int lane = threadIdx.x & 31;
    if (wave >= N) return;
    float4 h4 = ((const float4*)(h + (size_t)wave * DIM))[lane];
    float ss = h4.x * h4.x + h4.y * h4.y + h4.z * h4.z + h4.w * h4.w;
    #pragma unroll
    for (int off = 16; off; off >>= 1) ss += __shfl_xor(ss, off, 32);
    float inv = 1.0f / sqrtf(ss);
    float4 o;
    if (deg[wave] > 0) {
        o.x = h4.x * inv; o.y = h4.y * inv; o.z = h4.z * inv; o.w = h4.w * inv;
    } else {
        o = ((const float4*)(ent + (size_t)wave * DIM))[lane];
    }
    ((float4*)(out + (size_t)wave * DIM))[lane] = o;
}

// ---------- host ----------
extern "C" void kernel_launch(void* const* d_in, const int* in_sizes, int n_in,
                              void* d_out, int out_size, void* d_ws, size_t ws_size,
                              hipStream_t stream) {
    const float* ent  = (const float*)d_in[0];   // [N, D]
    const float* rel  = (const float*)d_in[1];   // [64, D]
    const float* W    = (const float*)d_in[2];   // [2, 256, 128]
    const float* bias = (const float*)d_in[3];   // [2, 128]
    const int* edst   = (const int*)d_in[4];
    const int* enbr   = (const int*)d_in[5];
    const int* erel   = (const int*)d_in[6];
    float* out = (float*)d_out;

    // workspace layout (all 16B aligned; sizes are multiples of 16B)
    char* ws = (char*)d_ws;
    size_t offND = (size_t)N_NODES * DIM * sizeof(float);       // 25.6 MB
    float*    prevA  = (float*)(ws);                            // ping buffer
    float*    neigh  = (float*)(ws + offND);
    float*    hbuf   = (float*)(ws + 2 * offND);
    float*    score  = (float*)(ws + 3 * offND);                // E floats
    unsigned* smax_u = (unsigned*)(ws + 3 * offND + (size_t)N_EDGES * 4);
    float*    denom  = (float*)((char*)smax_u + (size_t)N_NODES * 4);
    int*      deg    = (int*)((char*)denom + (size_t)N_NODES * 4);

    const int BT = 256;
    dim3 blk(BT);
    int gN   = (N_NODES + BT - 1) / BT;
    int gND  = (N_NODES * DIM + BT - 1) / BT;
    int gE   = (N_EDGES + BT - 1) / BT;
    int gEw  = (N_EDGES * 32 + BT - 1) / BT;     // wave per edge
    int gNw  = (N_NODES * 32 + BT - 1) / BT;     // wave per node
    int gMM  = (N_NODES / 16) * (DIM / 16) / (BT / 32); // 25000 waves / 8 = 3125

    // degree (layer-invariant)
    fill_u32_kernel<<<gN, blk, 0, stream>>>((unsigned*)deg, 0u, N_NODES);
    deg_kernel<<<gE, blk, 0, stream>>>(edst, deg, N_EDGES);

    for (int l = 0; l < N_LAYERS; ++l) {
        const float* prev = (l == 0) ? ent : prevA;
        float* nxt        = (l == N_LAYERS - 1) ? out : prevA;
        const float* Wl   = W + (size_t)l * 2 * DIM * DIM;
        const float* bl   = bias + (size_t)l * DIM;

        fill_u32_kernel<<<gN,  blk, 0, stream>>>(smax_u, 0u, N_NODES);
        fill_u32_kernel<<<gN,  blk, 0, stream>>>((unsigned*)denom, 0u, N_NODES);
        fill_u32_kernel<<<gND, blk, 0, stream>>>((unsigned*)neigh, 0u, N_NODES * DIM);

        score_kernel<<<gEw, blk, 0, stream>>>(prev, ent, rel, edst, enbr, erel,
                                              score, N_EDGES);
        smax_kernel<<<gE, blk, 0, stream>>>(score, edst, smax_u, N_EDGES);
        exdenom_kernel<<<gE, blk, 0, stream>>>(score, edst, smax_u, denom, N_EDGES);
        scatter_kernel<<<gEw, blk, 0, stream>>>(score, denom, prev, edst, enbr,
                                                neigh, N_EDGES);
        gemm_kernel<<<gMM, blk, 0, stream>>>(prev, neigh, Wl, bl, hbuf);
        norm_kernel<<<gNw, blk, 0, stream>>>(hbuf, deg, ent, nxt, N_NODES);
    }
}